// YukselSpline_19018115187078
// MI455X (gfx1250) — compile-verified
//
#include <hip/hip_runtime.h>

typedef __attribute__((ext_vector_type(2))) float v2f;
typedef __attribute__((ext_vector_type(8))) float v8f;

#define NPTS 18

// One 16x16 output tile per wave-job using V_WMMA_F32_16X16X4_F32.
// A (16x4 f32): lane l<16 holds M=l, K = 2*hi + v  (v = vector elem 0/1)
// B (4x16 f32): lane l<16 holds N=l, K = 2*hi + v  (mirror of A layout)
// C/D (16x16 f32): lane l holds N=l&15, M = v + 8*(l>>4)
// in_lds: [32][Kpad] row-major.  out_lds: [32][Npad] row-major.
// W: global (Nreal x Kreal) row-major; reads guarded so padded K/N contribute 0.
__device__ __forceinline__ void layer_wmma(
    const float* __restrict__ W, const float* __restrict__ B,
    const float* __restrict__ in_lds, float* __restrict__ out_lds,
    int Kreal, int Kpad, int Nreal, int Npad, int act,
    int wave, int nwaves, int lane)
{
  const int l  = lane & 15;
  const int hi = lane >> 4;
  const int ntiles = Npad >> 4;
  const int jobs   = ntiles * 2;          // 2 M-tiles (rows 0..31, valid 0..17)
  for (int j = wave; j < jobs; j += nwaves) {
    const int N0 = (j % ntiles) << 4;
    const int M0 = (j / ntiles) << 4;
    const int arow = (M0 + l) * Kpad;
    const int wn   = N0 + l;
    const bool wok = (wn < Nreal);
    v8f acc = {};
    for (int k = 0; k < Kpad; k += 4) {
      const int kk = k + 2 * hi;
      v2f a, b;
      a.x = in_lds[arow + kk];
      a.y = in_lds[arow + kk + 1];
      b.x = (wok && (kk     < Kreal)) ? W[wn * Kreal + kk]     : 0.0f;
      b.y = (wok && (kk + 1 < Kreal)) ? W[wn * Kreal + kk + 1] : 0.0f;
      acc = __builtin_amdgcn_wmma_f32_16x16x4_f32(false, a, false, b,
                                                  (short)0, acc, false, false);
    }
    const float bias = wok ? B[wn] : 0.0f;
#pragma unroll
    for (int v = 0; v < 8; ++v) {
      const int m = M0 + v + 8 * hi;
      if (m < NPTS) {
        float val = acc[v] + bias;
        if (act == 1)      val = 1.0f / (1.0f + expf(-val));   // sigmoid
        else if (act == 2) val = fmaxf(val, 0.0f);             // relu
        out_lds[m * Npad + wn] = val;
      }
    }
  }
}

// One-shot: MLP on pos=[1..18] -> P (18x4, cumsum over dims) -> 15-step
// control-point recurrence -> per-interval quadratic coefficients:
// entry i holds c0,c1,c2 (f_prev in d) and e0,e1,e2 (f_curr in d), each float4.
__global__ __launch_bounds__(256) void mlp_precompute(
    const float* __restrict__ W1, const float* __restrict__ b1,
    const float* __restrict__ W2, const float* __restrict__ b2,
    const float* __restrict__ W3, const float* __restrict__ b3,
    const float* __restrict__ W4, const float* __restrict__ b4,
    const float* __restrict__ W5, const float* __restrict__ b5,
    const float* __restrict__ W6, const float* __restrict__ b6,
    float* __restrict__ tbl)
{
  __shared__ float buf[2][32 * 256];
  const int tid  = threadIdx.x;
  const int lane = tid & 31;
  const int wave = tid >> 5;

  // Input: rows 0..17 col0 = pos (1..18), K padded to 4 with zeros.
  if (tid < 32 * 4) {
    const int m = tid >> 2, c = tid & 3;
    buf[0][tid] = (c == 0 && m < NPTS) ? (float)(m + 1) : 0.0f;
  }
  __syncthreads();
  layer_wmma(W1, b1, buf[0], buf[1],   1,   4,  16,  16, 1, wave, 8, lane); __syncthreads();
  layer_wmma(W2, b2, buf[1], buf[0],  16,  16,  64,  64, 1, wave, 8, lane); __syncthreads();
  layer_wmma(W3, b3, buf[0], buf[1],  64,  64, 256, 256, 2, wave, 8, lane); __syncthreads();
  layer_wmma(W4, b4, buf[1], buf[0], 256, 256,  64,  64, 2, wave, 8, lane); __syncthreads();
  layer_wmma(W5, b5, buf[0], buf[1],  64,  64,  16,  16, 2, wave, 8, lane); __syncthreads();
  layer_wmma(W6, b6, buf[1], buf[0],  16,  16,   4,  16, 0, wave, 8, lane); __syncthreads();

  if (tid == 0) {
    float P[NPTS][4];
    for (int m = 0; m < NPTS; ++m) {           // cumsum over dims (axis=1)
      float cum = 0.0f;
      for (int c = 0; c < 4; ++c) { cum += buf[0][m * 16 + c]; P[m][c] = cum; }
    }
    float T0[4], T1[4], T2[4];
    for (int c = 0; c < 4; ++c) { T0[c] = P[0][c]; T1[c] = P[1][c]; T2[c] = P[2][c]; }
    for (int i = 0; i < 15; ++i) {
      float n0[4], n1[4], n2[4];
      for (int c = 0; c < 4; ++c) {
        const float p0 = 0.25f * (T0[c] + T2[c]) + 0.5f * T1[c];
        const float p2 = P[3 + i][c];
        const float p1 = 2.0f * (T2[c] - 0.25f * (p0 + p2));
        // f_prev(d) = Bez(T, d+1/2): q0 + u*q1 + u^2*q2, u=d+1/2
        const float q1 = 2.0f * (T1[c] - T0[c]);
        const float q2 = T0[c] - 2.0f * T1[c] + T2[c];
        tbl[i * 24 +  0 + c] = T0[c] + 0.5f * q1 + 0.25f * q2;  // c0
        tbl[i * 24 +  4 + c] = q1 + q2;                         // c1
        tbl[i * 24 +  8 + c] = q2;                              // c2
        // f_curr(d) = Bez(Pnew, d)
        tbl[i * 24 + 12 + c] = p0;                              // e0
        tbl[i * 24 + 16 + c] = 2.0f * (p1 - p0);                // e1
        tbl[i * 24 + 20 + c] = p0 - 2.0f * p1 + p2;             // e2
        n0[c] = p0; n1[c] = p1; n2[c] = p2;
      }
      for (int c = 0; c < 4; ++c) { T0[c] = n0[c]; T1[c] = n1[c]; T2[c] = n2[c]; }
    }
  }
}

// HBM-bound streaming kernel: 4B in, 16B out per point (~160MB => ~7us @ 23.3TB/s).
__global__ __launch_bounds__(256) void spline_eval(
    const float* __restrict__ x, const float* __restrict__ tbl,
    float* __restrict__ out, int n)
{
  __shared__ float4 s[15 * 6];
  for (int t = threadIdx.x; t < 90; t += 256)
    s[t] = reinterpret_cast<const float4*>(tbl)[t];
  __syncthreads();

  const int idx = blockIdx.x * 256 + threadIdx.x;
  if (idx >= n) return;

  const float xv = x[idx];
  const float sc = xv * 15.0f;
  float fi = floorf(sc);
  fi = fminf(fmaxf(fi, 0.0f), 14.0f);     // last matching interval == floor bucket
  const float d = (sc - fi) * 0.5f;       // in [0, 0.5]
  const int base = (int)fi * 6;

  const float4 c0 = s[base + 0], c1 = s[base + 1], c2 = s[base + 2];
  const float4 e0 = s[base + 3], e1 = s[base + 4], e2 = s[base + 5];

  const float cp = __cosf(3.14159265358979323846f * d);
  const float w  = cp * cp;               // cos^2; sin^2 = 1 - w

  float4 r;
  {
    const float fp = fmaf(d, fmaf(d, c2.x, c1.x), c0.x);
    const float fc = fmaf(d, fmaf(d, e2.x, e1.x), e0.x);
    r.x = fmaf(w, fp - fc, fc);
  }
  {
    const float fp = fmaf(d, fmaf(d, c2.y, c1.y), c0.y);
    const float fc = fmaf(d, fmaf(d, e2.y, e1.y), e0.y);
    r.y = fmaf(w, fp - fc, fc);
  }
  {
    const float fp = fmaf(d, fmaf(d, c2.z, c1.z), c0.z);
    const float fc = fmaf(d, fmaf(d, e2.z, e1.z), e0.z);
    r.z = fmaf(w, fp - fc, fc);
  }
  {
    const float fp = fmaf(d, fmaf(d, c2.w, c1.w), c0.w);
    const float fc = fmaf(d, fmaf(d, e2.w, e1.w), e0.w);
    r.w = fmaf(w, fp - fc, fc);
  }
  reinterpret_cast<float4*>(out)[idx] = r;
}

extern "C" void kernel_launch(void* const* d_in, const int* in_sizes, int n_in,
                              void* d_out, int out_size, void* d_ws, size_t ws_size,
                              hipStream_t stream) {
  const float* x  = (const float*)d_in[0];
  const float* W1 = (const float*)d_in[1];
  const float* b1 = (const float*)d_in[2];
  const float* W2 = (const float*)d_in[3];
  const float* b2 = (const float*)d_in[4];
  const float* W3 = (const float*)d_in[5];
  const float* b3 = (const float*)d_in[6];
  const float* W4 = (const float*)d_in[7];
  const float* b4 = (const float*)d_in[8];
  const float* W5 = (const float*)d_in[9];
  const float* b5 = (const float*)d_in[10];
  const float* W6 = (const float*)d_in[11];
  const float* b6 = (const float*)d_in[12];
  float* out = (float*)d_out;
  float* tbl = (float*)d_ws;              // needs 15*24*4 = 1440 bytes
  const int n = in_sizes[0];

  mlp_precompute<<<1, 256, 0, stream>>>(W1, b1, W2, b2, W3, b3,
                                        W4, b4, W5, b5, W6, b6, tbl);
  const int blocks = (n + 255) / 256;
  spline_eval<<<blocks, 256, 0, stream>>>(x, tbl, out, n);
}